// Qfunction_32959579030378
// MI455X (gfx1250) — compile-verified
//
#include <hip/hip_runtime.h>
#include <math.h>

typedef float v2f __attribute__((ext_vector_type(2)));
typedef float v8f __attribute__((ext_vector_type(8)));

#define D      128
#define TILE   16
#define WAVES  4
#define AST    260   // padded LDS row stride (floats): 260 % 64 = 4 -> conflict-free frag reads

static __device__ __forceinline__ v8f wmma_f32_k4(v2f a, v2f b, v8f c) {
  // V_WMMA_F32_16X16X4_F32 : D = A(16x4) x B(4x16) + C(16x16), full fp32
  return __builtin_amdgcn_wmma_f32_16x16x4_f32(false, a, false, b, (short)0, c,
                                               false, false);
}

static __device__ __forceinline__ float sigmoidf(float x) {
  return 1.0f / (1.0f + __expf(-x));
}

// ---------------------------------------------------------------------------
// Kernel 1: seeds_mean[129] = sigmoid( (sum_s embed(seed_s) / seeds_num) @ W1 )
// Tiny (<=100 rows). Single block, plain VALU.
// ---------------------------------------------------------------------------
__global__ __launch_bounds__(256) void seeds_mean_kernel(
    const int* __restrict__ seeds, const float* __restrict__ snum_p,
    const int* __restrict__ all_nodes, const float* __restrict__ mask,
    const float* __restrict__ features, const int* __restrict__ neigh,
    const float* __restrict__ Wself, const float* __restrict__ Wneigh,
    const float* __restrict__ W1, float* __restrict__ smg,
    int S, int N, int K) {
  __shared__ float xs[D], agg[D], ssum[D];
  const int tid = threadIdx.x;
  if (tid < D) ssum[tid] = 0.0f;
  __syncthreads();
  for (int s = 0; s < S; ++s) {
    const int idx = seeds[s];          // uniform across block
    if (idx < N) {                     // idx == N is the zero pad row
      if (tid < D) {
        const int an = all_nodes[idx];
        xs[tid] = features[(size_t)an * D + tid];
        float a = 0.0f;
        for (int k = 0; k < K; ++k) {
          const int nb = neigh[(size_t)idx * K + k];
          a += features[(size_t)nb * D + tid];
        }
        agg[tid] = a / (float)K;
      }
      __syncthreads();
      if (tid < D) {
        float acc = 0.0f;
        for (int i = 0; i < D; ++i)
          acc += xs[i] * Wself[i * D + tid] + agg[i] * Wneigh[i * D + tid];
        ssum[tid] += fmaxf(acc, 0.0f) * mask[idx];
      }
      __syncthreads();
    }
  }
  const float sn = snum_p[0];
  if (tid <= D) {  // 129 outputs
    float h = 0.0f;
    for (int i = 0; i < D; ++i)
      h += (ssum[i] / sn) * W1[i * (D + 1) + tid];
    smg[tid] = sigmoidf(h);
  }
}

// ---------------------------------------------------------------------------
// Kernel 2: per 16-candidate tile (one per wave, 4 waves/block):
//   A = [x | neigh_mean]  (16 x 256, fp32, staged in LDS)
//   H = relu(A @ [Wself;Wneigh]) * mask        via v_wmma_f32_16x16x4_f32
//   P = sigmoid(H @ W3[:, 0:128])              via WMMA; col 128 scalar
//   Q = (1 - sum_j P*sm_j) * 100000
// ---------------------------------------------------------------------------
__global__ __launch_bounds__(WAVES * 32) void qfun_kernel(
    const int* __restrict__ cand, const int* __restrict__ all_nodes,
    const float* __restrict__ mask, const float* __restrict__ features,
    const int* __restrict__ neigh, const float* __restrict__ Wself,
    const float* __restrict__ Wneigh, const float* __restrict__ W3,
    const float* __restrict__ smg, float* __restrict__ out,
    int C, int K, float scale) {
  extern __shared__ float smem[];
  float* Alds  = smem;                        // WAVES*16*AST  (A_cat, later H)
  float* Wt    = Alds + WAVES * 16 * AST;     // 16*AST  B tile, [n][k] layout
  float* smld  = Wt + 16 * AST;               // 132
  float* w3c   = smld + 132;                  // 128   (W3 column 128)
  float* qbuf  = w3c + 128;                   // WAVES*16
  float* maskv = qbuf + WAVES * 16;           // WAVES*16
  int*   candI = (int*)(maskv + WAVES * 16);  // WAVES*16

  const int tid  = threadIdx.x;
  const int wave = tid >> 5;
  const int lane = tid & 31;
  const int lo   = lane & 15;   // N / M-low index within wave
  const int hi   = lane >> 4;   // selects K pair (A/B) and M half (C/D)
  const int blockBase = blockIdx.x * (WAVES * TILE);

  if (tid < WAVES * TILE) {
    const int c  = blockBase + tid;
    const int ci = (c < C) ? cand[c] : 0;     // clamp: keep EXEC all-ones later
    candI[tid] = ci;
    maskv[tid] = mask[ci];
  }
  if (tid < 132) smld[tid] = (tid < 129) ? smg[tid] : 0.0f;
  if (tid < 128) w3c[tid] = W3[tid * (D + 1) + D];
  __syncthreads();

  // ---- phase 1: build A_cat = [x | neigh_mean] per wave (16 x 256) ----
  float* Aw = Alds + wave * 16 * AST;
  for (int row = 0; row < TILE; ++row) {
    const int r  = candI[wave * TILE + row];
    const int an = all_nodes[r];
    const float4 v = *(const float4*)(features + (size_t)an * D + lane * 4);
    *(float4*)(Aw + row * AST + lane * 4) = v;
    float4 a = make_float4(0.f, 0.f, 0.f, 0.f);
    for (int k = 0; k < K; ++k) {
      const int nb = neigh[(size_t)r * K + k];
      const float4 t = *(const float4*)(features + (size_t)nb * D + lane * 4);
      a.x += t.x; a.y += t.y; a.z += t.z; a.w += t.w;
    }
    const float inv = 1.0f / (float)K;
    a.x *= inv; a.y *= inv; a.z *= inv; a.w *= inv;
    *(float4*)(Aw + row * AST + D + lane * 4) = a;
  }

  // ---- phase 2: H = relu(A_cat @ W_cat) * mask, K = 256 ----
  const int n16 = tid & 15;
  const int k0  = tid >> 4;  // 0..7
  v8f accH[8];
  for (int nt = 0; nt < 8; ++nt) {
    __syncthreads();
    {
      const int col = nt * 16 + n16;
      for (int i = 0; i < 16; ++i) {
        const int k = i * 8 + k0;
        Wt[n16 * AST + k]       = Wself[k * D + col];
        Wt[n16 * AST + 128 + k] = Wneigh[k * D + col];
      }
    }
    __syncthreads();
    v8f acc = {0.f, 0.f, 0.f, 0.f, 0.f, 0.f, 0.f, 0.f};
#pragma unroll 8
    for (int kk = 0; kk < 64; ++kk) {
      const int kI = 4 * kk + 2 * hi;
      const v2f a = *(const v2f*)(Aw + lo * AST + kI);
      const v2f b = *(const v2f*)(Wt + lo * AST + kI);
      acc = wmma_f32_k4(a, b, acc);
    }
    accH[nt] = acc;
  }

  // relu + mask; park H back into A region (cols 0..127), wave-local
  for (int nt = 0; nt < 8; ++nt) {
    const v8f acc = accH[nt];
    for (int v = 0; v < 8; ++v) {
      const int m = v + 8 * hi;
      float h = fmaxf(acc[v], 0.0f) * maskv[wave * TILE + m];
      Aw[m * AST + nt * 16 + lo] = h;
    }
  }

  // ---- phase 3: P = sigmoid(H @ W3[:,0:128]); accumulate P * sm ----
  float qp[8];
  for (int v = 0; v < 8; ++v) qp[v] = 0.0f;
  for (int nt = 0; nt < 8; ++nt) {
    __syncthreads();  // all waves done with previous Wt before overwrite
    {
      const int col = nt * 16 + n16;
      for (int i = 0; i < 16; ++i) {
        const int k = i * 8 + k0;
        Wt[n16 * AST + k] = W3[k * (D + 1) + col];
      }
    }
    __syncthreads();
    v8f acc = {0.f, 0.f, 0.f, 0.f, 0.f, 0.f, 0.f, 0.f};
#pragma unroll 8
    for (int kk = 0; kk < 32; ++kk) {
      const int kI = 4 * kk + 2 * hi;
      const v2f a = *(const v2f*)(Aw + lo * AST + kI);
      const v2f b = *(const v2f*)(Wt + lo * AST + kI);
      acc = wmma_f32_k4(a, b, acc);
    }
    const float w = smld[nt * 16 + lo];
    for (int v = 0; v < 8; ++v) qp[v] += sigmoidf(acc[v]) * w;
  }

  // reduce over the 16 N-lanes (butterfly inside each half-wave)
  for (int off = 1; off < 16; off <<= 1)
    for (int v = 0; v < 8; ++v) qp[v] += __shfl_xor(qp[v], off, 32);
  if (lo == 0)
    for (int v = 0; v < 8; ++v) qbuf[wave * TILE + v + 8 * hi] = qp[v];
  __syncthreads();

  // column 128 of W3 (scalar per row) + final Q
  if (lane < 16) {
    const int m = lane;
    const int c = blockBase + wave * TILE + m;
    if (c < C) {
      float d = 0.0f;
      for (int i = 0; i < D; ++i) d += Aw[m * AST + i] * w3c[i];
      const float p = sigmoidf(d) * smld[D];
      const float q = qbuf[wave * TILE + m] + p;
      out[c] = (1.0f - q) * scale;
    }
  }
}

extern "C" void kernel_launch(void* const* d_in, const int* in_sizes, int n_in,
                              void* d_out, int out_size, void* d_ws, size_t ws_size,
                              hipStream_t stream) {
  const int*   seeds    = (const int*)d_in[0];
  const float* snum     = (const float*)d_in[1];
  const int*   cand     = (const int*)d_in[2];
  const int*   allnodes = (const int*)d_in[3];
  const float* mask     = (const float*)d_in[4];
  const float* features = (const float*)d_in[5];
  const int*   neigh    = (const int*)d_in[6];
  const float* Wself    = (const float*)d_in[7];
  const float* Wneigh   = (const float*)d_in[8];
  const float* W1       = (const float*)d_in[9];
  // d_in[10] = W2: feeds only the unused global_mean -> skipped
  const float* W3       = (const float*)d_in[11];

  const int S = in_sizes[0];
  const int C = in_sizes[2];
  const int N = in_sizes[3];
  const int K = in_sizes[6] / N;

  float* smg = (float*)d_ws;  // seeds_mean[129] scratch
  (void)n_in; (void)out_size; (void)ws_size;

  seeds_mean_kernel<<<1, 256, 0, stream>>>(seeds, snum, allnodes, mask, features,
                                           neigh, Wself, Wneigh, W1, smg, S, N, K);

  const int blocks = (C + WAVES * TILE - 1) / (WAVES * TILE);
  const size_t shmem =
      (size_t)(WAVES * 16 * AST + 16 * AST + 132 + 128 + WAVES * 16 + WAVES * 16 +
               WAVES * 16) * sizeof(float);  // ~85 KB, well under 320 KB/WGP
  qfun_kernel<<<blocks, WAVES * 32, shmem, stream>>>(
      cand, allnodes, mask, features, neigh, Wself, Wneigh, W3, smg,
      (float*)d_out, C, K, 100000.0f /* GRAPH_SIZE */);
}